// ifftLayer_42880953483789
// MI455X (gfx1250) — compile-verified
//
#include <hip/hip_runtime.h>

typedef float v2f __attribute__((ext_vector_type(2)));
typedef float v8f __attribute__((ext_vector_type(8)));
typedef int   v4i __attribute__((ext_vector_type(4)));

typedef __attribute__((address_space(1))) v4i* gas_v4i;   // global (AS1)
typedef __attribute__((address_space(3))) v4i* las_v4i;   // LDS (AS3)

#define NN   256
#define MAT  (NN * NN)          // 65536 elements per plane per batch
#define NBAT 256

#define LSTRIDE 20              // left LDS row stride: 80B (16B aligned, conflict-free)

#if defined(__HIP_DEVICE_COMPILE__) && __has_builtin(__builtin_amdgcn_global_load_async_to_lds_b128)
#define USE_ASYNC_LDS 1
#else
#define USE_ASYNC_LDS 0
#endif

// ---------------------------------------------------------------------------
// Kernel 0: centered-IFFT twiddle planes.
//   A[m,k] = (1/256) * exp(+2*pi*i * (((m+128)*(k+128)) mod 256) / 256)
// Ar = cos/256, Ai = sin/256, Ain = -sin/256 (pre-negated plane: f32 WMMA NEG
// bits only negate C, so -Li*Ri is realized by a negated operand).
// ---------------------------------------------------------------------------
__global__ __launch_bounds__(256) void build_twiddle(float* __restrict__ Ar,
                                                     float* __restrict__ Ai,
                                                     float* __restrict__ Ain) {
    const int m = blockIdx.x;
    const int k = threadIdx.x;
    const int p = (((m + 128) & 255) * ((k + 128) & 255)) & 255;
    const float ang = 6.28318530717958647692f * (float)p * (1.0f / 256.0f);
    float s, c;
    __sincosf(ang, &s, &c);
    const float scale = 1.0f / 256.0f;
    const int idx = m * NN + k;
    Ar[idx]  =  c * scale;
    Ai[idx]  =  s * scale;
    Ain[idx] = -s * scale;
}

// ---------------------------------------------------------------------------
// helpers
// ---------------------------------------------------------------------------
__device__ __forceinline__ void wait_async0() {
#if USE_ASYNC_LDS
#if __has_builtin(__builtin_amdgcn_s_wait_asynccnt)
    __builtin_amdgcn_s_wait_asynccnt(0);
#else
    asm volatile("s_wait_asynccnt 0" ::: "memory");
#endif
#endif
}

#if USE_ASYNC_LDS
__device__ __forceinline__ void async_cp16(const float* gsrc, float* ldst) {
    __builtin_amdgcn_global_load_async_to_lds_b128((gas_v4i)gsrc, (las_v4i)ldst, 0, 0);
}
#endif

// One K-chunk (16 wide) of the 4-real-GEMM complex product:
//   accR += Lr*Rr + Mr*Mi ; accI += Lr*Ri + Li*Rr
__device__ __forceinline__ void compute_chunk(
    const float* __restrict__ sL0, const float* __restrict__ sL1,
    const float* __restrict__ sL2,
    const float* __restrict__ sR0, const float* __restrict__ sR1,
    const float* __restrict__ sR2,
    v8f accR[2], v8f accI[2],
    int rowSub, int colHalf, int ml, int khalf)
{
    for (int kk = 0; kk < 16; kk += 4) {
        const int kb    = kk + (khalf << 1);            // K pair base, this half-wave
        const int lbase = (rowSub + ml) * LSTRIDE + kb; // even -> 8B aligned

        const v2f flr = *(const v2f*)&sL0[lbase];
        const v2f fli = *(const v2f*)&sL1[lbase];
        const v2f fmr = *(const v2f*)&sL2[lbase];

        for (int c = 0; c < 2; ++c) {
            const int n = colHalf + c * 16 + ml;
            v2f frr, fri, fmi;
            frr[0] = sR0[kb * 64 + n]; frr[1] = sR0[(kb + 1) * 64 + n];
            fri[0] = sR1[kb * 64 + n]; fri[1] = sR1[(kb + 1) * 64 + n];
            fmi[0] = sR2[kb * 64 + n]; fmi[1] = sR2[(kb + 1) * 64 + n];

            accR[c] = __builtin_amdgcn_wmma_f32_16x16x4_f32(
                false, flr, false, frr, (short)0, accR[c], false, false);
            accR[c] = __builtin_amdgcn_wmma_f32_16x16x4_f32(
                false, fmr, false, fmi, (short)0, accR[c], false, false);
            accI[c] = __builtin_amdgcn_wmma_f32_16x16x4_f32(
                false, flr, false, fri, (short)0, accI[c], false, false);
            accI[c] = __builtin_amdgcn_wmma_f32_16x16x4_f32(
                false, fli, false, frr, (short)0, accI[c], false, false);
        }
    }
}

// ---------------------------------------------------------------------------
// Planar complex GEMM C = L * R via v_wmma_f32_16x16x4_f32.
// 64x64 output tile / block, 8 wave32s, LDS double-buffered async staging.
// ---------------------------------------------------------------------------
__global__ __launch_bounds__(256) void cgemm_wmma_f32(
    const float* __restrict__ Lr, const float* __restrict__ Li,
    const float* __restrict__ Mr,
    const float* __restrict__ Rr, const float* __restrict__ Ri,
    const float* __restrict__ Mi,
    long long sLr, long long sLi, long long sMr,
    long long sRr, long long sRi, long long sMi,
    float* __restrict__ Cr, float* __restrict__ Ci)
{
    __shared__ __align__(16) float sL[2][3][64 * LSTRIDE]; // 30 KB
    __shared__ __align__(16) float sR[2][3][16 * 64];      // 24 KB

    const int tid     = threadIdx.x;
    const int lane    = tid & 31;
    const int w       = tid >> 5;
    const int rowSub  = (w & 3) * 16;
    const int colHalf = (w >> 2) * 32;
    const int ml      = lane & 15;
    const int khalf   = lane >> 4;
    const int b       = blockIdx.z;
    const int rowTile = blockIdx.y * 64;
    const int colTile = blockIdx.x * 64;

    const float* Lp[3] = { Lr + (long long)b * sLr,
                           Li + (long long)b * sLi,
                           Mr + (long long)b * sMr };
    const float* Rp[3] = { Rr + (long long)b * sRr,
                           Ri + (long long)b * sRi,
                           Mi + (long long)b * sMi };

    // per-thread copy slots: left 64x16 tile as 256 x 16B chunks, right 16x64 same
    const int lrow = tid >> 2,  lc4 = (tid & 3) * 4;
    const int rk   = tid >> 4,  rn4 = (tid & 15) * 4;

    v8f accR[2] = {};
    v8f accI[2] = {};

#if USE_ASYNC_LDS
    // ---- async double-buffered pipeline ----
    for (int p = 0; p < 3; ++p) {   // prologue: chunk 0 into buffer 0
        async_cp16(Lp[p] + (long long)(rowTile + lrow) * NN + lc4,
                   &sL[0][p][lrow * LSTRIDE + lc4]);
        async_cp16(Rp[p] + (long long)rk * NN + colTile + rn4,
                   &sR[0][p][rk * 64 + rn4]);
    }
    int cur = 0;
    for (int k0 = 0; k0 < NN; k0 += 16) {
        wait_async0();       // own copies into buf[cur] done
        __syncthreads();     // everyone's copies done; buf[cur^1] fully consumed
        const int nxt = cur ^ 1;
        if (k0 + 16 < NN) {
            const int k1 = k0 + 16;
            for (int p = 0; p < 3; ++p) {
                async_cp16(Lp[p] + (long long)(rowTile + lrow) * NN + k1 + lc4,
                           &sL[nxt][p][lrow * LSTRIDE + lc4]);
                async_cp16(Rp[p] + (long long)(k1 + rk) * NN + colTile + rn4,
                           &sR[nxt][p][rk * 64 + rn4]);
            }
        }
        compute_chunk(sL[cur][0], sL[cur][1], sL[cur][2],
                      sR[cur][0], sR[cur][1], sR[cur][2],
                      accR, accI, rowSub, colHalf, ml, khalf);
        cur = nxt;
    }
#else
    // ---- fallback: synchronous staging, single buffer ----
    for (int k0 = 0; k0 < NN; k0 += 16) {
        for (int p = 0; p < 3; ++p) {
            const float* gl = Lp[p] + (long long)(rowTile + lrow) * NN + k0 + lc4;
            const float* gr = Rp[p] + (long long)(k0 + rk) * NN + colTile + rn4;
            for (int q = 0; q < 4; ++q) {
                sL[0][p][lrow * LSTRIDE + lc4 + q] = gl[q];
                sR[0][p][rk * 64 + rn4 + q]        = gr[q];
            }
        }
        __syncthreads();
        compute_chunk(sL[0][0], sL[0][1], sL[0][2],
                      sR[0][0], sR[0][1], sR[0][2],
                      accR, accI, rowSub, colHalf, ml, khalf);
        __syncthreads();
    }
#endif

    // epilogue: C/D 16x16 f32 layout (VGPR j -> row j / 8+j)
    const int rhalf = khalf * 8;
    for (int c = 0; c < 2; ++c) {
        for (int j = 0; j < 8; ++j) {
            const int row = rowTile + rowSub + rhalf + j;
            const int col = colTile + colHalf + c * 16 + ml;
            const long long off = (long long)b * MAT + (long long)row * NN + col;
            Cr[off] = accR[c][j];
            Ci[off] = accI[c][j];
        }
    }
}

// ---------------------------------------------------------------------------
// Launch: twiddles, then T = A*X, then Y = T*A.  T (134 MB) lives in d_ws and
// stays resident in the 192 MB L2, so HBM traffic ~ read X + write Y = 268 MB
// -> ~11.5 us at 23.3 TB/s; f32 WMMA compute (~69 GFLOP) is ~50x below that.
// ---------------------------------------------------------------------------
extern "C" void kernel_launch(void* const* d_in, const int* in_sizes, int n_in,
                              void* d_out, int out_size, void* d_ws, size_t ws_size,
                              hipStream_t stream) {
    const float* xr = (const float*)d_in[0];
    const float* xi = (const float*)d_in[1];

    float* Ar  = (float*)d_ws;
    float* Ai  = Ar + MAT;
    float* Ain = Ai + MAT;
    float* Tr  = Ain + MAT;
    float* Ti  = Tr + (long long)NBAT * MAT;

    float* Yr = (float*)d_out;
    float* Yi = Yr + (long long)NBAT * MAT;

    build_twiddle<<<NN, NN, 0, stream>>>(Ar, Ai, Ain);

    dim3 grid(4, 4, NBAT);
    // Stage 1: T = A * X   (accR = Ar*Xr + Ain*Xi ; accI = Ar*Xi + Ai*Xr)
    cgemm_wmma_f32<<<grid, 256, 0, stream>>>(
        Ar, Ai, Ain, xr, xi, xi,
        0LL, 0LL, 0LL, (long long)MAT, (long long)MAT, (long long)MAT,
        Tr, Ti);

    // Stage 2: Y = T * A   (accR = Tr*Ar + Ti*Ain ; accI = Tr*Ai + Ti*Ar)
    cgemm_wmma_f32<<<grid, 256, 0, stream>>>(
        Tr, Ti, Ti, Ar, Ai, Ain,
        (long long)MAT, (long long)MAT, (long long)MAT, 0LL, 0LL, 0LL,
        Yr, Yi);
}